// ResdualAttentionBlock_23596550324700
// MI455X (gfx1250) — compile-verified
//
#include <hip/hip_runtime.h>
#include <hip/hip_bf16.h>

// ---------------- problem constants ----------------
#define N_   4
#define C_   512
#define T_   4096
#define H_   8
#define DH_  64
#define CH_  64              // chunk size
#define NC_  (T_ / CH_)      // 64 chunks
#define NTC  ((size_t)N_ * T_ * C_)   // elements of one [N,T,C] tensor
#define SCALE 0.125f         // 1/sqrt(64)
#define EPSV  1e-5f
#define NEGV  (-1e9f)
#define LD   68              // padded LDS row stride (words): 16B-aligned,
                             // bank = (4*row + k) % 64 -> conflict-free frags

typedef __attribute__((ext_vector_type(2))) float v2f;
typedef __attribute__((ext_vector_type(8))) float v8f;

// D = A(16x4) * B(4x16) + C, full fp32 (V_WMMA_F32_16X16X4_F32)
__device__ __forceinline__ v8f wmma_f32(v2f a, v2f b, v8f c) {
  return __builtin_amdgcn_wmma_f32_16x16x4_f32(
      /*neg_a=*/false, a, /*neg_b=*/false, b,
      /*c_mod=*/(short)0, c, /*reuse_a=*/false, /*reuse_b=*/false);
}

__device__ __forceinline__ v8f v8zero() {
  v8f z = {0.f, 0.f, 0.f, 0.f, 0.f, 0.f, 0.f, 0.f};
  return z;
}

// =====================================================================
// Kernel 1: fused Q/K/V projection.
//   Out[mat][n][t][co] = sum_c in_mat[n][c][t] * W_in[mat*C+co][c] + b
// Block = 64(t) x 64(co) tile of one matrix; W K-chunks staged in padded
// LDS once per block (coalesced float4), shared by all 4 waves.
// A-fragment loads stay global: lane = t -> fully coalesced.
// =====================================================================
__global__ __launch_bounds__(128) void k_proj(
    const float* __restrict__ q, const float* __restrict__ k,
    const float* __restrict__ v, const float* __restrict__ W,
    const float* __restrict__ b, float* __restrict__ ws) {
  __shared__ float Wl[64 * LD];        // W chunk: [co 0..63][k 0..63]

  const int tid  = threadIdx.x;
  const int lane = tid & 31;
  const int wv   = tid >> 5;
  const int m    = lane & 15;
  const int hi   = (lane >> 4) << 1;   // K sub-offset: 0 or 2
  const int roff = (lane >> 4) << 3;   // C/D row offset: 0 or 8

  const int mat = blockIdx.x >> 11;    // 2048 blocks per matrix
  const int rem = blockIdx.x & 2047;
  const int co0 = (rem & 7) << 6;      // output-channel tile base
  const int r0  = (rem >> 3) << 6;     // 64-row block over N*T
  const int n   = r0 >> 12;            // / T_
  const int tb  = r0 & (T_ - 1);
  const int t0  = tb + (wv << 4);      // wave's 16-row strip

  const float* in = (mat == 0) ? q : (mat == 1) ? k : v;
  in += (size_t)n * C_ * T_;
  const float* Wm = W + (size_t)(mat * C_ + co0) * C_;

  v8f acc[4];
#pragma unroll
  for (int j = 0; j < 4; ++j) {
    float bb = b[mat * C_ + co0 + 16 * j + m];
    v8f t = {bb, bb, bb, bb, bb, bb, bb, bb};
    acc[j] = t;
  }

  for (int kb = 0; kb < C_; kb += 64) {
    // stage W[co0..co0+63][kb..kb+63] (coalesced float4 per row)
    for (int idx = tid; idx < 64 * 16; idx += 128) {
      const int r = idx >> 4, c4 = (idx & 15) << 2;
      *(float4*)&Wl[r * LD + c4] = *(const float4*)&Wm[(size_t)r * C_ + kb + c4];
    }
    __syncthreads();
#pragma unroll 4
    for (int k0 = 0; k0 < 64; k0 += 4) {
      const int kk = k0 + hi;
      v2f a;
      a.x = in[(size_t)(kb + kk) * T_ + t0 + m];
      a.y = in[(size_t)(kb + kk + 1) * T_ + t0 + m];
#pragma unroll
      for (int j = 0; j < 4; ++j) {
        const v2f bf = *(const v2f*)&Wl[(16 * j + m) * LD + kk];
        acc[j] = wmma_f32(a, bf, acc[j]);
      }
    }
    __syncthreads();
  }

  float* out = ws + (size_t)mat * NTC + (size_t)(n * T_ + t0) * C_ + co0;
#pragma unroll
  for (int j = 0; j < 4; ++j)
#pragma unroll
    for (int i = 0; i < 8; ++i)
      out[(size_t)(i + roff) * C_ + 16 * j + m] = acc[j][i];
}

// =====================================================================
// Kernel 2: intra-chunk attention per (n, chunk, head).
//   S = scale*(Qh Kh^T) + mask ; P = softmax(S) ; O = P Vh
// All LDS tiles padded to LD words/row -> conflict-free ds_load_b64.
// O tile overwrites its own Q tile in ws (all Q reads happen first).
// =====================================================================
__global__ __launch_bounds__(128) void k_attn(
    const float* __restrict__ masks, float* __restrict__ ws) {
  __shared__ float Qs[64 * LD];   // Q tile, later reused for V tile
  __shared__ float Ks[64 * LD];
  __shared__ float Ss[64 * LD];   // scores -> probabilities

  const int n     = blockIdx.x >> 9;
  const int chunk = (blockIdx.x >> 3) & 63;
  const int h     = blockIdx.x & 7;
  const int lane  = threadIdx.x & 31;
  const int m     = lane & 15;
  const int hi    = (lane >> 4) << 1;
  const int roff  = (lane >> 4) << 3;
  const int m0    = (threadIdx.x >> 5) << 4;   // wave's 16-row strip

  const size_t tileBase = (size_t)(n * T_ + chunk * CH_) * C_ + h * DH_;
  const float* Qg = ws + tileBase;               // region 0
  const float* Kg = ws + NTC + tileBase;         // region 1
  const float* Vg = ws + 2 * NTC + tileBase;     // region 2

  // stage Q and K tiles (64 x 64 f32) into padded LDS
  for (int idx = threadIdx.x; idx < 64 * 16; idx += 128) {
    const int r = idx >> 4, c4 = (idx & 15) << 2;
    *(float4*)&Qs[r * LD + c4] = *(const float4*)&Qg[(size_t)r * C_ + c4];
    *(float4*)&Ks[r * LD + c4] = *(const float4*)&Kg[(size_t)r * C_ + c4];
  }
  __syncthreads();

  // S strip = Q[m0..m0+15][:] * K^T   (contraction over Dh=64)
  v8f acc[4];
#pragma unroll
  for (int j = 0; j < 4; ++j) acc[j] = v8zero();
#pragma unroll 4
  for (int k0 = 0; k0 < DH_; k0 += 4) {
    const int kk = k0 + hi;
    const v2f a = *(const v2f*)&Qs[(m0 + m) * LD + kk];
#pragma unroll
    for (int j = 0; j < 4; ++j) {
      const v2f bf = *(const v2f*)&Ks[(16 * j + m) * LD + kk];
      acc[j] = wmma_f32(a, bf, acc[j]);
    }
  }
  // scale + key-validity mask, spill strip to LDS
  const float* mk = masks + (size_t)n * T_ + chunk * CH_;
#pragma unroll
  for (int j = 0; j < 4; ++j) {
    const int col = 16 * j + m;
    const float madd = (mk[col] > 0.f) ? 0.f : NEGV;
#pragma unroll
    for (int i = 0; i < 8; ++i)
      Ss[(m0 + i + roff) * LD + col] = acc[j][i] * SCALE + madd;
  }
  __syncthreads();

  // stage V tile into Qs (Q no longer needed); softmax rows in parallel
  for (int idx = threadIdx.x; idx < 64 * 16; idx += 128) {
    const int r = idx >> 4, c4 = (idx & 15) << 2;
    *(float4*)&Qs[r * LD + c4] = *(const float4*)&Vg[(size_t)r * C_ + c4];
  }
  if (threadIdx.x < 64) {
    float* row = &Ss[threadIdx.x * LD];
    float mx = row[0];
    for (int j = 1; j < 64; ++j) mx = fmaxf(mx, row[j]);
    float sum = 0.f;
    for (int j = 0; j < 64; ++j) { float e = __expf(row[j] - mx); row[j] = e; sum += e; }
    const float inv = 1.f / sum;
    for (int j = 0; j < 64; ++j) row[j] *= inv;
  }
  __syncthreads();

  // O strip = P[m0..m0+15][:] * V   (contraction over keys)
  v8f o[4];
#pragma unroll
  for (int j = 0; j < 4; ++j) o[j] = v8zero();
#pragma unroll 4
  for (int k0 = 0; k0 < CH_; k0 += 4) {
    const int kk = k0 + hi;
    const v2f a = *(const v2f*)&Ss[(m0 + m) * LD + kk];
#pragma unroll
    for (int j = 0; j < 4; ++j) {
      const int dcol = 16 * j + m;               // Dh index
      v2f bf;
      bf.x = Qs[kk * LD + dcol];
      bf.y = Qs[(kk + 1) * LD + dcol];
      o[j] = wmma_f32(a, bf, o[j]);
    }
  }
  // store O over the Q tile (region 0) -> becomes AttnOut [N,T,C]
  float* Og = ws + tileBase + (size_t)m0 * C_;
#pragma unroll
  for (int j = 0; j < 4; ++j)
#pragma unroll
    for (int i = 0; i < 8; ++i)
      Og[(size_t)(i + roff) * C_ + 16 * j + m] = o[j][i];
}

// =====================================================================
// Kernel 3: out-projection + mask + residual, stored transposed [N,C,T].
//   Y[n][c][t] = (sum_j Wout[c][j]*Attn[n][t][j] + b_out[c])*mask + x[n][c][t]
// Block = 64(c) x 64(t); both Wout and Attn K-chunks staged in padded LDS.
// =====================================================================
__global__ __launch_bounds__(128) void k_outproj(
    const float* __restrict__ Wout, const float* __restrict__ bout,
    const float* __restrict__ x, const float* __restrict__ masks,
    float* __restrict__ ws) {
  __shared__ float Wl[64 * LD];   // Wout chunk: [c 0..63][k 0..63]
  __shared__ float Al[64 * LD];   // Attn chunk: [t 0..63][k 0..63]

  const int tid  = threadIdx.x;
  const int lane = tid & 31;
  const int wv   = tid >> 5;
  const int m    = lane & 15;
  const int hi   = (lane >> 4) << 1;
  const int roff = (lane >> 4) << 3;

  const int n   = blockIdx.x >> 9;           // 512 blocks per batch
  const int rem = blockIdx.x & 511;
  const int cb  = (rem >> 6) << 6;           // 8 c-tiles of 64
  const int tb  = (rem & 63) << 6;           // 64 t-tiles of 64
  const int c0  = cb + (wv << 4);            // wave's 16-row c strip

  const float* attn = ws;                    // region 0 (AttnOut [N,T,C])
  float*       Y    = ws + NTC;              // region 1 reused for [N,C,T]

  v8f acc[4];
#pragma unroll
  for (int j = 0; j < 4; ++j) {
    acc[j] = v8zero();
#pragma unroll
    for (int i = 0; i < 8; ++i) acc[j][i] = bout[c0 + i + roff];
  }

  for (int kb = 0; kb < C_; kb += 64) {
    for (int idx = tid; idx < 64 * 16; idx += 128) {
      const int r = idx >> 4, c4 = (idx & 15) << 2;
      *(float4*)&Wl[r * LD + c4] =
          *(const float4*)&Wout[(size_t)(cb + r) * C_ + kb + c4];
      *(float4*)&Al[r * LD + c4] =
          *(const float4*)&attn[(size_t)(n * T_ + tb + r) * C_ + kb + c4];
    }
    __syncthreads();
#pragma unroll 4
    for (int k0 = 0; k0 < 64; k0 += 4) {
      const int kk = k0 + hi;
      const v2f a = *(const v2f*)&Wl[((wv << 4) + m) * LD + kk];
#pragma unroll
      for (int j = 0; j < 4; ++j) {
        const v2f bf = *(const v2f*)&Al[(16 * j + m) * LD + kk];
        acc[j] = wmma_f32(a, bf, acc[j]);
      }
    }
    __syncthreads();
  }

#pragma unroll
  for (int j = 0; j < 4; ++j)
#pragma unroll
    for (int i = 0; i < 8; ++i) {
      const int c = c0 + i + roff;
      const int t = tb + 16 * j + m;
      const float mval = masks[(size_t)n * T_ + t];
      const size_t off = (size_t)(n * C_ + c) * T_ + t;
      Y[off] = acc[j][i] * mval + x[off];
    }
}

// =====================================================================
// Kernel 4: InstanceNorm1d over T per (n,c), two-pass for accuracy.
// =====================================================================
__global__ __launch_bounds__(256) void k_inorm(
    const float* __restrict__ Y, float* __restrict__ out) {
  __shared__ float red[256];
  const int row = blockIdx.x;                 // n*C + c
  const int tid = threadIdx.x;
  const float* yr = Y + (size_t)row * T_;

  float s = 0.f;
  for (int t = tid; t < T_; t += 256) s += yr[t];
  red[tid] = s;
  __syncthreads();
  for (int off = 128; off > 0; off >>= 1) {
    if (tid < off) red[tid] += red[tid + off];
    __syncthreads();
  }
  const float mu = red[0] * (1.f / T_);
  __syncthreads();

  float s2 = 0.f;
  for (int t = tid; t < T_; t += 256) { const float d = yr[t] - mu; s2 += d * d; }
  red[tid] = s2;
  __syncthreads();
  for (int off = 128; off > 0; off >>= 1) {
    if (tid < off) red[tid] += red[tid + off];
    __syncthreads();
  }
  const float inv = rsqrtf(red[0] * (1.f / T_) + EPSV);

  float* orow = out + (size_t)row * T_;
  for (int t = tid; t < T_; t += 256) orow[t] = (yr[t] - mu) * inv;
}

// =====================================================================
extern "C" void kernel_launch(void* const* d_in, const int* in_sizes, int n_in,
                              void* d_out, int out_size, void* d_ws, size_t ws_size,
                              hipStream_t stream) {
  (void)in_sizes; (void)n_in; (void)out_size; (void)ws_size;
  const float* x     = (const float*)d_in[0];
  const float* q     = (const float*)d_in[1];
  const float* k     = (const float*)d_in[2];
  const float* v     = (const float*)d_in[3];
  const float* masks = (const float*)d_in[4];
  const float* W_in  = (const float*)d_in[5];
  const float* b_in  = (const float*)d_in[6];
  const float* W_out = (const float*)d_in[7];
  const float* b_out = (const float*)d_in[8];
  float* out = (float*)d_out;
  float* ws  = (float*)d_ws;   // needs 3 * N*T*C floats = 96 MB

  // 1) Q/K/V projections: 3 * 256 * 8 = 6144 blocks (64t x 64co tiles)
  k_proj<<<6144, 128, 0, stream>>>(q, k, v, W_in, b_in, ws);
  // 2) chunk attention: one block per (n, chunk, head)
  k_attn<<<N_ * NC_ * H_, 128, 0, stream>>>(masks, ws);
  // 3) out-projection + residual -> [N,C,T] in region 1
  k_outproj<<<2048, 128, 0, stream>>>(W_out, b_out, x, masks, ws);
  // 4) instance norm -> d_out
  k_inorm<<<N_ * C_, 256, 0, stream>>>(ws + NTC, out);
}